// STGformer_56564719288824
// MI455X (gfx1250) — compile-verified
//
#include <hip/hip_runtime.h>
#include <hip/hip_bf16.h>

typedef __attribute__((ext_vector_type(2))) float v2f;
typedef __attribute__((ext_vector_type(8))) float v8f;

#define GF_RELU   1
#define GF_BIAS   2
#define GF_RESID  4

__device__ __forceinline__ float wave_sum32(float v) {
#pragma unroll
  for (int o = 16; o > 0; o >>= 1) v += __shfl_xor(v, o, 32);
  return v;
}
__device__ __forceinline__ float wave_max32(float v) {
#pragma unroll
  for (int o = 16; o > 0; o >>= 1) v = fmaxf(v, __shfl_xor(v, o, 32));
  return v;
}

// ------------------------------------------------------------------
// fp32 WMMA GEMM: C = epilogue(A[M,K] @ B[K,N]).
// One 32x32 C tile per wave (4 accumulators), 8 waves per block.
// K stepped by 4 with V_WMMA_F32_16X16X4_F32, software-pipelined.
// OOB lanes are handled by CLAMPING load addresses (results discarded
// at store time) so the inner loop has no predication at all.
// K must be a multiple of 4 (true for all uses here).
// A fragment (16x4, ISA 7.12.2): lanes 0-15 -> M=lane, v0=K0,v1=K1;
//                                lanes 16-31 -> M=lane-16, v0=K2,v1=K3.
// B fragment (4x16): row-striped across lanes (mirror of A/C layout).
// C fragment (16x16): vgpr r: lanes0-15 M=r, lanes16-31 M=r+8, N=lane&15.
// ------------------------------------------------------------------
template <int TB>
__global__ __launch_bounds__(256) void gemm_f32_wmma(
    const float* __restrict__ A, int lda, long long sA,
    const float* __restrict__ B, int ldb, long long sB,
    float* __restrict__ C, int ldc, long long sC,
    const float* __restrict__ bias,
    const float* __restrict__ resid, long long sR,
    int M, int N, int K, int flags)
{
  int batch = blockIdx.y;
  A += (long long)batch * sA;
  B += (long long)batch * sB;
  C += (long long)batch * sC;
  if (resid) resid += (long long)batch * sR;

  int lane = threadIdx.x & 31;
  int wave = threadIdx.x >> 5;
  int tilesN = (N + 31) >> 5;
  int tilesM = (M + 31) >> 5;
  int tile = blockIdx.x * 8 + wave;
  if (tile >= tilesM * tilesN) return;
  int tm = tile / tilesN, tn = tile % tilesN;

  int half = lane >> 4;   // 0: holds K+0/K+1 ; 1: holds K+2/K+3
  int l16  = lane & 15;
  int am0 = tm * 32 + l16;        // rows of upper 16x32 sub-tile
  int am1 = am0 + 16;             // rows of lower 16x32 sub-tile
  int bn0 = tn * 32 + l16;        // cols of left 32x16 sub-tile
  int bn1 = bn0 + 16;             // cols of right 32x16 sub-tile
  int am0c = min(am0, M - 1), am1c = min(am1, M - 1);
  int bn0c = min(bn0, N - 1), bn1c = min(bn1, N - 1);

  // Streaming cursors (contiguous b64-able pairs for A and TB-B).
  const float* pA0 = A + (long long)am0c * lda + half * 2;
  const float* pA1 = A + (long long)am1c * lda + half * 2;
  const float* pB0;
  const float* pB1;
  long long bstep;
  if (TB) {  // B element (k,n) at B[n*ldb + k]
    pB0 = B + (long long)bn0c * ldb + half * 2;
    pB1 = B + (long long)bn1c * ldb + half * 2;
    bstep = 4;
  } else {   // B element (k,n) at B[k*ldb + n]
    pB0 = B + (long long)(half * 2) * ldb + bn0c;
    pB1 = B + (long long)(half * 2) * ldb + bn1c;
    bstep = 4LL * ldb;
  }

  v8f acc00 = {0.f,0.f,0.f,0.f,0.f,0.f,0.f,0.f};
  v8f acc01 = acc00, acc10 = acc00, acc11 = acc00;

  v2f a0, a1, b0, b1;
  // prologue load (k = 0)
  a0.x = pA0[0]; a0.y = pA0[1];
  a1.x = pA1[0]; a1.y = pA1[1];
  if (TB) { b0.x = pB0[0]; b0.y = pB0[1]; b1.x = pB1[0]; b1.y = pB1[1]; }
  else    { b0.x = pB0[0]; b0.y = pB0[ldb]; b1.x = pB1[0]; b1.y = pB1[ldb]; }
  pA0 += 4; pA1 += 4; pB0 += bstep; pB1 += bstep;

  for (int k0 = 4; k0 < K; k0 += 4) {
    v2f na0, na1, nb0, nb1;
    na0.x = pA0[0]; na0.y = pA0[1];
    na1.x = pA1[0]; na1.y = pA1[1];
    if (TB) { nb0.x = pB0[0]; nb0.y = pB0[1]; nb1.x = pB1[0]; nb1.y = pB1[1]; }
    else    { nb0.x = pB0[0]; nb0.y = pB0[ldb]; nb1.x = pB1[0]; nb1.y = pB1[ldb]; }
    pA0 += 4; pA1 += 4; pB0 += bstep; pB1 += bstep;

    acc00 = __builtin_amdgcn_wmma_f32_16x16x4_f32(false, a0, false, b0, (short)0, acc00, false, false);
    acc01 = __builtin_amdgcn_wmma_f32_16x16x4_f32(false, a0, false, b1, (short)0, acc01, false, false);
    acc10 = __builtin_amdgcn_wmma_f32_16x16x4_f32(false, a1, false, b0, (short)0, acc10, false, false);
    acc11 = __builtin_amdgcn_wmma_f32_16x16x4_f32(false, a1, false, b1, (short)0, acc11, false, false);

    a0 = na0; a1 = na1; b0 = nb0; b1 = nb1;
  }
  acc00 = __builtin_amdgcn_wmma_f32_16x16x4_f32(false, a0, false, b0, (short)0, acc00, false, false);
  acc01 = __builtin_amdgcn_wmma_f32_16x16x4_f32(false, a0, false, b1, (short)0, acc01, false, false);
  acc10 = __builtin_amdgcn_wmma_f32_16x16x4_f32(false, a1, false, b0, (short)0, acc10, false, false);
  acc11 = __builtin_amdgcn_wmma_f32_16x16x4_f32(false, a1, false, b1, (short)0, acc11, false, false);

  // Epilogue: store valid rows/cols only.
  auto store_tile = [&](const v8f& acc, int rowBase, int col) {
    if (col >= N) return;
    float bv = (flags & GF_BIAS) ? bias[col] : 0.f;
#pragma unroll
    for (int r = 0; r < 8; ++r) {
      int row = rowBase + r;
      if (row < M) {
        float vv = acc[r] + bv;
        if (flags & GF_RESID) vv += resid[(long long)row * ldc + col];
        if (flags & GF_RELU)  vv = fmaxf(vv, 0.f);
        C[(long long)row * ldc + col] = vv;
      }
    }
  };
  int rb = tm * 32 + half * 8;
  store_tile(acc00, rb,      bn0);
  store_tile(acc01, rb,      bn1);
  store_tile(acc10, rb + 16, bn0);
  store_tile(acc11, rb + 16, bn1);
}

// ------------------------------------------------------------------
// X0[t*N+n, 0:24]  = x[n,t,:] @ W_in + b_in   (K=3, pointwise)
// X0[t*N+n, 24:104]= adp[t,n,:]
// ------------------------------------------------------------------
__global__ __launch_bounds__(256) void build_x0_kernel(
    const float* __restrict__ x, const float* __restrict__ W_in,
    const float* __restrict__ b_in, const float* __restrict__ adp,
    float* __restrict__ X0, int T, int Nn)
{
  long long idx = (long long)blockIdx.x * blockDim.x + threadIdx.x;
  long long total = (long long)T * Nn * 104;
  if (idx >= total) return;
  int d = (int)(idx % 104);
  long long row = idx / 104;
  int n = (int)(row % Nn);
  int t = (int)(row / Nn);
  float o;
  if (d < 24) {
    const float* xp = x + ((long long)n * T + t) * 3;
    o = b_in[d] + xp[0] * W_in[d] + xp[1] * W_in[24 + d] + xp[2] * W_in[48 + d];
  } else {
    o = adp[((long long)t * Nn + n) * 80 + (d - 24)];
  }
  X0[idx] = o;
}

// Row softmax (input already ReLU'd by GEMM epilogue). Wave per row.
__global__ __launch_bounds__(256) void softmax_rows_kernel(
    float* __restrict__ g, long long rows, int ncols)
{
  int lane = threadIdx.x & 31, wave = threadIdx.x >> 5;
  long long row = (long long)blockIdx.x * 8 + wave;
  if (row >= rows) return;
  float* p = g + row * (long long)ncols;
  float m = -1e30f;
  for (int c = lane; c < ncols; c += 32) m = fmaxf(m, p[c]);
  m = wave_max32(m);
  float s = 0.f;
  for (int c = lane; c < ncols; c += 32) s += __expf(p[c] - m);
  s = wave_sum32(s);
  float inv = 1.f / s;
  for (int c = lane; c < ncols; c += 32) p[c] = __expf(p[c] - m) * inv;
}

// ------------------------------------------------------------------
// Kernelized linear attention for one (outer, head).
// spatial=1: outer=t in [0,T), sequence l over n (L=Nn), rows bt=outer*Nn+l
// spatial=0: outer=n in [0,Nn), sequence l over t (L=T),  rows bt=l*Nn+outer
// QKV row layout: [q(104) | k(104) | v(104)], head h at offset h*26.
// Writes OUT[bt, (spatial?0:104) + h*26 + d]  (CAT buffer, 208 cols).
// ------------------------------------------------------------------
#define LCHUNK 64
__global__ __launch_bounds__(256) void lin_att_kernel(
    const float* __restrict__ QKV, float* __restrict__ OUT,
    int T, int Nn, int spatial)
{
  const int HD = 26;
  __shared__ float kn_s[LCHUNK][HD + 1];
  __shared__ float v_s [LCHUNK][HD + 1];
  __shared__ float kvs_s[HD][HD + 1];
  __shared__ float ksum_s[HD];

  int h = blockIdx.x & 3;
  int outer = blockIdx.x >> 2;
  int L = spatial ? Nn : T;
  int tid = threadIdx.x;

  float acc0 = 0.f, acc1 = 0.f, acc2 = 0.f;  // 676 kvs outputs / 256 threads
  float ksacc = 0.f;

  for (int l0 = 0; l0 < L; l0 += LCHUNK) {
    int lc = min(LCHUNK, L - l0);
    for (int e = tid; e < lc * HD; e += 256) {
      int r = e / HD, d = e % HD;
      long long bt = spatial ? ((long long)outer * Nn + (l0 + r))
                             : ((long long)(l0 + r) * Nn + outer);
      const float* rp = QKV + bt * 312 + h * HD;
      kn_s[r][d] = rp[104 + d];
      v_s [r][d] = rp[208 + d];
    }
    __syncthreads();
    if (tid < lc) {  // normalize k rows: kn = k / max(||k||, 1e-12)
      float s = 0.f;
#pragma unroll
      for (int d = 0; d < HD; ++d) { float t = kn_s[tid][d]; s += t * t; }
      float inv = 1.f / fmaxf(sqrtf(s), 1e-12f);
#pragma unroll
      for (int d = 0; d < HD; ++d) kn_s[tid][d] *= inv;
    }
    __syncthreads();
    {
      int p = tid;
      if (p < 676) { int m = p / HD, d = p % HD; float a = acc0;
        for (int r = 0; r < lc; ++r) a += kn_s[r][m] * v_s[r][d]; acc0 = a; }
      p = tid + 256;
      if (p < 676) { int m = p / HD, d = p % HD; float a = acc1;
        for (int r = 0; r < lc; ++r) a += kn_s[r][m] * v_s[r][d]; acc1 = a; }
      p = tid + 512;
      if (p < 676) { int m = p / HD, d = p % HD; float a = acc2;
        for (int r = 0; r < lc; ++r) a += kn_s[r][m] * v_s[r][d]; acc2 = a; }
      if (tid < HD) { float s = ksacc;
        for (int r = 0; r < lc; ++r) s += kn_s[r][tid]; ksacc = s; }
    }
    __syncthreads();
  }
  { int p = tid;       if (p < 676) kvs_s[p / HD][p % HD] = acc0; }
  { int p = tid + 256; if (p < 676) kvs_s[p / HD][p % HD] = acc1; }
  { int p = tid + 512; if (p < 676) kvs_s[p / HD][p % HD] = acc2; }
  if (tid < HD) ksum_s[tid] = ksacc;
  __syncthreads();

  int lane = tid & 31, wave = tid >> 5;
  float Lf = (float)L;
  int outCol = (spatial ? 0 : 104) + h * HD;
  for (int l = wave; l < L; l += 8) {
    long long bt = spatial ? ((long long)outer * Nn + l)
                           : ((long long)l * Nn + outer);
    const float* rp = QKV + bt * 312 + h * HD;
    float q = (lane < HD) ? rp[lane] : 0.f;
    float nrm = wave_sum32(q * q);
    float inv = 1.f / fmaxf(sqrtf(nrm), 1e-12f);
    float qn = q * inv;
    float num = 0.f;
#pragma unroll
    for (int m = 0; m < HD; ++m) {
      float qm = __shfl(qn, m, 32);  // broadcast qn[m] from lane m
      if (lane < HD) num += qm * kvs_s[m][lane];
    }
    float dp = (lane < HD) ? qn * ksum_s[lane] : 0.f;
    float den = wave_sum32(dp) + Lf;
    if (lane < HD) {
      float vv = rp[208 + lane];
      OUT[bt * 208 + outCol + lane] = (num + Lf * vv) / den;
    }
  }
}

// g = H + att0*P0 + 0.01*att1*P1 ; out = LayerNorm(2*g). Wave per row, D=104.
__global__ __launch_bounds__(256) void fuse_g_ln1_kernel(
    const float* __restrict__ H, const float* __restrict__ att0,
    const float* __restrict__ P0, const float* __restrict__ att1,
    const float* __restrict__ P1, const float* __restrict__ gamma,
    const float* __restrict__ beta, float* __restrict__ out, long long rows)
{
  const int D = 104;
  int lane = threadIdx.x & 31, wave = threadIdx.x >> 5;
  long long row = (long long)blockIdx.x * 8 + wave;
  if (row >= rows) return;
  long long base = row * D;
  float v[4]; float s = 0.f;
#pragma unroll
  for (int j = 0; j < 4; ++j) {
    int c = lane + 32 * j; float t = 0.f;
    if (c < D) {
      long long i = base + c;
      t = 2.f * (H[i] + att0[i] * P0[i] + 0.01f * att1[i] * P1[i]);
    }
    v[j] = t; s += t;
  }
  float mean = wave_sum32(s) * (1.0f / 104.0f);
  float var = 0.f;
#pragma unroll
  for (int j = 0; j < 4; ++j) {
    int c = lane + 32 * j;
    if (c < D) { float d2 = v[j] - mean; var += d2 * d2; }
  }
  var = wave_sum32(var) * (1.0f / 104.0f);
  float rstd = rsqrtf(var + 1e-5f);
#pragma unroll
  for (int j = 0; j < 4; ++j) {
    int c = lane + 32 * j;
    if (c < D) out[base + c] = (v[j] - mean) * rstd * gamma[c] + beta[c];
  }
}

// In-place LayerNorm over rows of 104.
__global__ __launch_bounds__(256) void ln_inplace_kernel(
    float* __restrict__ io, const float* __restrict__ gamma,
    const float* __restrict__ beta, long long rows)
{
  const int D = 104;
  int lane = threadIdx.x & 31, wave = threadIdx.x >> 5;
  long long row = (long long)blockIdx.x * 8 + wave;
  if (row >= rows) return;
  long long base = row * D;
  float v[4]; float s = 0.f;
#pragma unroll
  for (int j = 0; j < 4; ++j) {
    int c = lane + 32 * j; float t = 0.f;
    if (c < D) t = io[base + c];
    v[j] = t; s += t;
  }
  float mean = wave_sum32(s) * (1.0f / 104.0f);
  float var = 0.f;
#pragma unroll
  for (int j = 0; j < 4; ++j) {
    int c = lane + 32 * j;
    if (c < D) { float d2 = v[j] - mean; var += d2 * d2; }
  }
  var = wave_sum32(var) * (1.0f / 104.0f);
  float rstd = rsqrtf(var + 1e-5f);
#pragma unroll
  for (int j = 0; j < 4; ++j) {
    int c = lane + 32 * j;
    if (c < D) io[base + c] = (v[j] - mean) * rstd * gamma[c] + beta[c];
  }
}

// h2t[n, t*104+d] = h2[(t*Nn+n)*104 + d]
__global__ __launch_bounds__(256) void transpose_h2_kernel(
    const float* __restrict__ h2, float* __restrict__ h2t, int T, int Nn)
{
  const int D = 104;
  long long idx = (long long)blockIdx.x * blockDim.x + threadIdx.x;
  long long total = (long long)Nn * T * D;
  if (idx >= total) return;
  int n = (int)(idx / ((long long)T * D));
  int r = (int)(idx % ((long long)T * D));
  int t = r / D, d = r % D;
  h2t[idx] = h2[((long long)t * Nn + n) * D + d];
}

// ------------------------------------------------------------------
static inline void gemm(hipStream_t st,
                        const float* A, int lda, long long sA,
                        const float* B, int ldb, long long sB,
                        float* C, int ldc, long long sC,
                        const float* bias, const float* resid, long long sR,
                        int M, int N, int K, int flags, int batches, int transB)
{
  int tiles = ((M + 31) / 32) * ((N + 31) / 32);
  dim3 grid((unsigned)((tiles + 7) / 8), (unsigned)batches, 1);
  if (transB)
    gemm_f32_wmma<1><<<grid, 256, 0, st>>>(A, lda, sA, B, ldb, sB, C, ldc, sC,
                                           bias, resid, sR, M, N, K, flags);
  else
    gemm_f32_wmma<0><<<grid, 256, 0, st>>>(A, lda, sA, B, ldb, sB, C, ldc, sC,
                                           bias, resid, sR, M, N, K, flags);
}

extern "C" void kernel_launch(void* const* d_in, const int* in_sizes, int n_in,
                              void* d_out, int out_size, void* d_ws, size_t ws_size,
                              hipStream_t stream) {
  (void)in_sizes; (void)n_in; (void)out_size; (void)ws_size;
  const int T = 365, Nn = 400, D = 104;
  const long long BT = (long long)T * Nn;          // 146000
  const long long BTD  = BT * D;                   // 15,184,000
  const long long BT2D = BT * 2 * D;               // 30,368,000
  const long long BT3D = BT * 3 * D;               // 45,552,000

  const float* x      = (const float*)d_in[0];
  const float* W_in   = (const float*)d_in[1];
  const float* b_in   = (const float*)d_in[2];
  const float* adp    = (const float*)d_in[3];
  const float* W_tp   = (const float*)d_in[4];
  const float* b_tp   = (const float*)d_in[5];
  const float* qkv_w  = (const float*)d_in[6];   // [2,104,312]
  const float* op_w   = (const float*)d_in[7];   // [2,208,104]
  const float* op_b   = (const float*)d_in[8];   // [2,104]
  const float* pw_w   = (const float*)d_in[9];   // [2,104,104]
  const float* pw_b   = (const float*)d_in[10];  // [2,104]
  const float* fc_w1  = (const float*)d_in[11];  // [104,208]
  const float* fc_b1  = (const float*)d_in[12];
  const float* fc_w2  = (const float*)d_in[13];  // [208,104]
  const float* fc_b2  = (const float*)d_in[14];
  const float* ln1_g  = (const float*)d_in[15];
  const float* ln1_b  = (const float*)d_in[16];
  const float* ln2_g  = (const float*)d_in[17];
  const float* ln2_b  = (const float*)d_in[18];
  const float* ep_w   = (const float*)d_in[19];  // [37960,104]
  const float* ep_b   = (const float*)d_in[20];
  const float* enc_w1 = (const float*)d_in[21];  // [3,104,208]
  const float* enc_b1 = (const float*)d_in[22];  // [3,208]
  const float* enc_w2 = (const float*)d_in[23];  // [3,208,104]
  const float* enc_b2 = (const float*)d_in[24];  // [3,104]
  const float* out_w  = (const float*)d_in[25];  // [104,365]
  const float* out_b  = (const float*)d_in[26];

  // Workspace layout (floats), regions aliased by lifetime (~781 MB total):
  float* ws = (float*)d_ws;
  float* H     = ws;                     // [BT,104] live through step 8
  float* X0    = ws + BTD;               // [BT,104]; later att0, then h2
  float* att0  = X0;
  float* h2    = X0;
  float* GR    = ws + 2 * BTD;           // graph [365,400,400] = 58.4M floats
  float* graph = GR;
  float* CAT0  = GR;                     // [BT,208] (graph dead by then)
  float* att1  = GR + BT2D;              // [BT,104]
  float* h2t   = GR;                     // [400,37960] (CAT0 dead by then)
  float* Z1    = ws + 2 * BTD + 58400000LL;  // [BT,104]; later P0
  float* P0    = Z1;
  float* QKV0  = Z1 + BTD;               // [BT,312]; later CAT1 + P1
  float* CAT1  = QKV0;
  float* P1    = QKV0 + BT2D;
  float* QKV1  = QKV0 + BT3D;            // [BT,312]; later h1 + M1
  float* h1    = QKV1;
  float* M1    = QKV1 + BTD;
  float* y     = QKV1 + BT3D;            // [400,104]
  float* yt    = y + 400 * 208;          // [400,208]

  const unsigned ROWBLK = (unsigned)((BT + 7) / 8);      // 18250 (8 rows/block)
  const unsigned ELBLK  = (unsigned)((BTD + 255) / 256); // 59313

  // S0: build X0 = [x@W_in+b_in | adp]
  build_x0_kernel<<<ELBLK, 256, 0, stream>>>(x, W_in, b_in, adp, X0, T, Nn);
  // S1: H = X0 @ W_tp + b_tp
  gemm(stream, X0, D, 0, W_tp, D, 0, H, D, 0, b_tp, nullptr, 0,
       (int)BT, D, D, GF_BIAS, 1, 0);
  // S2: graph[t] = relu(adp[t] @ adp[t]^T)   (batched, B transposed)
  gemm(stream, adp, 80, 32000, adp, 80, 32000, graph, 400, 160000,
       nullptr, nullptr, 0, 400, 400, 80, GF_RELU, T, 1);
  softmax_rows_kernel<<<ROWBLK, 256, 0, stream>>>(graph, BT, 400);
  // S3: Z1[t] = graph[t] @ H[t]
  gemm(stream, graph, 400, 160000, H, D, (long long)400 * D,
       Z1, D, (long long)400 * D, nullptr, nullptr, 0, 400, D, 400, 0, T, 0);
  // S4: QKV projections (no bias in reference)
  gemm(stream, H, D, 0, qkv_w, 312, 0, QKV0, 312, 0, nullptr, nullptr, 0,
       (int)BT, 312, D, 0, 1, 0);
  gemm(stream, Z1, D, 0, qkv_w + 104 * 312, 312, 0, QKV1, 312, 0,
       nullptr, nullptr, 0, (int)BT, 312, D, 0, 1, 0);
  // S5: linear attention (spatial fills cols 0..103, temporal 104..207)
  lin_att_kernel<<<(unsigned)(T * 4), 256, 0, stream>>>(QKV0, CAT0, T, Nn, 1);
  lin_att_kernel<<<(unsigned)(Nn * 4), 256, 0, stream>>>(QKV0, CAT0, T, Nn, 0);
  lin_att_kernel<<<(unsigned)(T * 4), 256, 0, stream>>>(QKV1, CAT1, T, Nn, 1);
  lin_att_kernel<<<(unsigned)(Nn * 4), 256, 0, stream>>>(QKV1, CAT1, T, Nn, 0);
  // S6: attention output projections
  gemm(stream, CAT0, 208, 0, op_w, D, 0, att0, D, 0, op_b, nullptr, 0,
       (int)BT, D, 208, GF_BIAS, 1, 0);
  gemm(stream, CAT1, 208, 0, op_w + 208 * 104, D, 0, att1, D, 0,
       op_b + 104, nullptr, 0, (int)BT, D, 208, GF_BIAS, 1, 0);
  // S7: pointwise gates  P0 = H@pw_w0+b,  P1 = att0@pw_w1+b
  gemm(stream, H, D, 0, pw_w, D, 0, P0, D, 0, pw_b, nullptr, 0,
       (int)BT, D, D, GF_BIAS, 1, 0);
  gemm(stream, att0, D, 0, pw_w + 104 * 104, D, 0, P1, D, 0,
       pw_b + 104, nullptr, 0, (int)BT, D, D, GF_BIAS, 1, 0);
  // S8: g = H + att0*P0 + 0.01*att1*P1 ; h1 = LN1(2g)
  fuse_g_ln1_kernel<<<ROWBLK, 256, 0, stream>>>(H, att0, P0, att1, P1,
                                                ln1_g, ln1_b, h1, BT);
  // S9/S10: MLP with residual, then LN2 -> h2
  gemm(stream, h1, D, 0, fc_w1, 208, 0, M1, 208, 0, fc_b1, nullptr, 0,
       (int)BT, 208, D, GF_BIAS | GF_RELU, 1, 0);
  gemm(stream, M1, 208, 0, fc_w2, D, 0, h2, D, 0, fc_b2, h1, 0,
       (int)BT, D, 208, GF_BIAS | GF_RESID, 1, 0);
  ln_inplace_kernel<<<ROWBLK, 256, 0, stream>>>(h2, ln2_g, ln2_b, BT);
  // S11/S12: encoder projection over flattened (t,d)
  transpose_h2_kernel<<<ELBLK, 256, 0, stream>>>(h2, h2t, T, Nn);
  gemm(stream, h2t, T * D, 0, ep_w, D, 0, y, D, 0, ep_b, nullptr, 0,
       Nn, D, T * D, GF_BIAS, 1, 0);
  // S13: three residual MLP blocks on y [400,104]
  for (int i = 0; i < 3; ++i) {
    gemm(stream, y, D, 0, enc_w1 + (long long)i * 104 * 208, 208, 0,
         yt, 208, 0, enc_b1 + i * 208, nullptr, 0,
         Nn, 208, D, GF_BIAS | GF_RELU, 1, 0);
    gemm(stream, yt, 208, 0, enc_w2 + (long long)i * 208 * 104, D, 0,
         y, D, 0, enc_b2 + i * 104, y, 0,
         Nn, D, 208, GF_BIAS | GF_RESID, 1, 0);
  }
  // S14: out = y @ out_w + out_b  -> [400,365] == d_out (N,T,1) row-major
  gemm(stream, y, D, 0, out_w, T, 0, (float*)d_out, T, 0, out_b, nullptr, 0,
       Nn, T, D, GF_BIAS, 1, 0);
}